// SpectralConv_31731218383013
// MI455X (gfx1250) — compile-verified
//
#include <hip/hip_runtime.h>
#include <hip/hip_bf16.h>

typedef __attribute__((ext_vector_type(2))) float v2f;
typedef __attribute__((ext_vector_type(8))) float v8f;

namespace {
constexpr int kN    = 10242;            // icosphere level-5 nodes
constexpr int kNNZ  = 71694;            // edges
constexpr int kBT   = 32;               // B*T
constexpr int kC    = 16;               // in channels
constexpr int kCout = 32;               // out channels
constexpr int kKs   = 20;               // Chebyshev order
constexpr int kF    = kC * kBT;         // 512 feature columns per node
constexpr int kCK   = kC * kKs;         // 320 (W inner dim)
constexpr int kNTiles = (kN + 15) / 16; // 641 (last tile partial)
}

// x[bt, n, c] -> t0[n, bt*C + c]  (column order within a node row is arbitrary for the
// SpMM; [bt*C + c] makes each lane's two WMMA K-values contiguous -> b64 loads)
__global__ void transpose_kernel(const float* __restrict__ x, float* __restrict__ t0) {
  int tid = blockIdx.x * blockDim.x + threadIdx.x;
  if (tid >= kN * kF) return;
  int c  = tid & (kC - 1);
  int bt = (tid >> 4) & (kBT - 1);
  int n  = tid >> 9;
  t0[tid] = x[((size_t)bt * kN + n) * kC + c];
}

// W[co, c*Ks + k] -> Wp[((k*8 + c/2)*32 + co)*2 + (c&1)]  : per-lane-contiguous B pairs
__global__ void pack_w_kernel(const float* __restrict__ W, float* __restrict__ Wp) {
  int tid = blockIdx.x * blockDim.x + threadIdx.x;
  if (tid >= kKs * kC * kCout) return;
  int co = tid & (kCout - 1);
  int c  = (tid >> 5) & (kC - 1);
  int k  = tid >> 9;
  Wp[(((size_t)(k * 8 + (c >> 1)) * kCout + co) << 1) | (c & 1)] =
      W[(size_t)co * kCK + c * kKs + k];
}

__global__ void zero_kernel(float* __restrict__ p, int count) {
  int tid = blockIdx.x * blockDim.x + threadIdx.x;
  if (tid < count) p[tid] = 0.0f;
}

// dst = mul * src  (folds the "- T_{k-2}" of the recurrence into the scatter target)
__global__ void scaled_copy_kernel(float* __restrict__ dst, const float* __restrict__ src,
                                   float mul, int count) {
  int tid = blockIdx.x * blockDim.x + threadIdx.x;
  if (tid < count) dst[tid] = mul * src[tid];
}

// dst[row, :] += scale * val * src[col, :]; one thread per (edge, 4-col chunk)
__global__ void spmm_scatter_kernel(float* __restrict__ dst, const float* __restrict__ src,
                                    const int* __restrict__ rows, const int* __restrict__ cols,
                                    const float* __restrict__ vals, float scale) {
  int tid = blockIdx.x * blockDim.x + threadIdx.x;
  int e = tid >> 7;                 // kF/4 = 128 chunks per edge
  int f = (tid & 127) << 2;
  if (e >= kNNZ) return;
  float v = vals[e] * scale;
  int col = cols[e];
  int row = rows[e];
  float4 s = *reinterpret_cast<const float4*>(src + (size_t)col * kF + f);
  float* d = dst + (size_t)row * kF + f;
  atomicAdd(d + 0, v * s.x);
  atomicAdd(d + 1, v * s.y);
  atomicAdd(d + 2, v * s.z);
  atomicAdd(d + 3, v * s.w);
}

// One wave computes out[bt, n0:n0+16, 0:32] += Tk_tile(16x16) @ Wk(16x32) for one term k:
// 8 chained V_WMMA_F32_16X16X4_F32 (two Cout tiles share the A registers).
// Accumulation into out via global_atomic_add_f32 (no RMW load latency; each location
// touched by exactly one lane per launch, launches stream-ordered -> deterministic).
__global__ void acc_wmma_kernel(float* __restrict__ out, const float* __restrict__ tk,
                                const float* __restrict__ Wp, int k) {
  int wid  = (blockIdx.x * blockDim.x + threadIdx.x) >> 5; // wave-uniform
  int lane = threadIdx.x & 31;
  if (wid >= kNTiles * kBT) return;       // wave-uniform: EXEC all-ones through WMMA

  int bt = wid & (kBT - 1);
  int nt = wid >> 5;
  int n0 = nt * 16;

  int m  = lane & 15;   // A row / B-and-D column within tile
  int kg = lane >> 4;   // half-wave group

  int nr = n0 + m;                        // clamp partial tile: garbage rows never stored
  if (nr >= kN) nr = kN - 1;
  const v2f* arow = reinterpret_cast<const v2f*>(tk + (size_t)nr * kF + (size_t)bt * kC);
  const v2f* wk   = reinterpret_cast<const v2f*>(Wp) + (size_t)k * 8 * kCout;

  v8f d0 = {};
  v8f d1 = {};
#pragma unroll
  for (int kb = 0; kb < kC; kb += 4) {
    // ISA 16x4 f32 A layout: lanes 0-15 hold K=0,1; lanes 16-31 hold K=2,3
    int cpair = (kb >> 1) + kg;           // (c0, c0+1) pair index, c0 = kb + 2*kg
    v2f a  = arow[cpair];                 // global_load_b64
    v2f b0 = wk[(size_t)cpair * kCout + m];
    v2f b1 = wk[(size_t)cpair * kCout + 16 + m];
    d0 = __builtin_amdgcn_wmma_f32_16x16x4_f32(false, a, false, b0, (short)0, d0,
                                               false, false);
    d1 = __builtin_amdgcn_wmma_f32_16x16x4_f32(false, a, false, b1, (short)0, d1,
                                               false, false);
  }

  // C/D layout: VGPR v -> row v (lanes 0-15) or v+8 (lanes 16-31), col = lane%16
  float* obase = out + ((size_t)bt * kN + n0) * kCout + m;
  if (n0 + 16 <= kN) {                    // 640 of 641 tiles: no per-element branching
#pragma unroll
    for (int v = 0; v < 8; ++v) {
      size_t o = (size_t)(v + kg * 8) * kCout;
      atomicAdd(obase + o,      d0[v]);
      atomicAdd(obase + o + 16, d1[v]);
    }
  } else {                                // last partial tile only
#pragma unroll
    for (int v = 0; v < 8; ++v) {
      int n = n0 + v + kg * 8;
      if (n < kN) {
        size_t o = (size_t)(v + kg * 8) * kCout;
        atomicAdd(obase + o,      d0[v]);
        atomicAdd(obase + o + 16, d1[v]);
      }
    }
  }
}

extern "C" void kernel_launch(void* const* d_in, const int* in_sizes, int n_in,
                              void* d_out, int out_size, void* d_ws, size_t ws_size,
                              hipStream_t stream) {
  (void)in_sizes; (void)n_in; (void)out_size; (void)ws_size;

  const float* x    = (const float*)d_in[0];
  const int*   erow = (const int*)d_in[1];
  const int*   ecol = (const int*)d_in[2];
  const float* eval = (const float*)d_in[3];
  const float* W    = (const float*)d_in[4];
  float* out = (float*)d_out;

  // Ring of 3 Chebyshev-term buffers (21 MB each, L2-resident) + 40 KB packed W.
  float* Ta = (float*)d_ws;
  float* Tb = Ta + (size_t)kN * kF;
  float* Tc = Tb + (size_t)kN * kF;
  float* Wp = Tc + (size_t)kN * kF;

  const int  NF = kN * kF;
  const dim3 blk(256);
  auto nblk = [](long n) { return dim3((unsigned)((n + 255) / 256)); };
  const long scThreads  = (long)kNNZ * (kF / 4);
  const long accThreads = (long)kNTiles * kBT * 32;

  // Prep: T0 = transposed input; Wp packed; out = 0; out += proj(T0, k=0)
  transpose_kernel<<<nblk(NF), blk, 0, stream>>>(x, Ta);
  pack_w_kernel<<<nblk((long)kKs * kC * kCout), blk, 0, stream>>>(W, Wp);
  zero_kernel<<<nblk((long)kBT * kN * kCout), blk, 0, stream>>>(out, kBT * kN * kCout);
  acc_wmma_kernel<<<nblk(accThreads), blk, 0, stream>>>(out, Ta, Wp, 0);

  // T1 = L @ T0
  zero_kernel<<<nblk(NF), blk, 0, stream>>>(Tb, NF);
  spmm_scatter_kernel<<<nblk(scThreads), blk, 0, stream>>>(Tb, Ta, erow, ecol, eval, 1.0f);
  acc_wmma_kernel<<<nblk(accThreads), blk, 0, stream>>>(out, Tb, Wp, 1);

  // T_k = 2 * L @ T_{k-1} - T_{k-2}
  for (int k = 2; k < kKs; ++k) {
    scaled_copy_kernel<<<nblk(NF), blk, 0, stream>>>(Tc, Ta, -1.0f, NF);
    spmm_scatter_kernel<<<nblk(scThreads), blk, 0, stream>>>(Tc, Tb, erow, ecol, eval, 2.0f);
    acc_wmma_kernel<<<nblk(accThreads), blk, 0, stream>>>(out, Tc, Wp, k);
    float* tmp = Ta; Ta = Tb; Tb = Tc; Tc = tmp;
  }
}